// CorotationalBeam2DNormalized_42734924595225
// MI455X (gfx1250) — compile-verified
//
#include <hip/hip_runtime.h>

#define EPSF 1e-10f
#define TILE 256

struct F3 { float x, y, z; };

typedef int v2i __attribute__((vector_size(8)));
typedef __attribute__((address_space(1))) int  gint_t;
typedef __attribute__((address_space(3))) int  lint_t;
typedef __attribute__((address_space(1))) v2i  gv2i_t;
typedef __attribute__((address_space(3))) v2i  lv2i_t;

// ---------------------------------------------------------------------------
// CDNA5 async Global->LDS helpers (ASYNCcnt path).
// ---------------------------------------------------------------------------
__device__ __forceinline__ void async_ld_b32(const void* g, void* l) {
    __builtin_amdgcn_global_load_async_to_lds_b32(
        (gint_t*)g, (lint_t*)l, 0, 0);
}

__device__ __forceinline__ void async_ld_b64(const void* g, void* l) {
    __builtin_amdgcn_global_load_async_to_lds_b64(
        (gv2i_t*)g, (lv2i_t*)l, 0, 0);
}

#define WAIT_ASYNC(n) do { __builtin_amdgcn_s_wait_asynccnt(n); \
                           asm volatile("" ::: "memory"); } while (0)

__device__ __forceinline__ void atomic_add_f32(float* p, float v) {
    // Relaxed, device-scope -> non-returning global_atomic_add_f32 (STOREcnt).
    __hip_atomic_fetch_add(p, v, __ATOMIC_RELAXED, __HIP_MEMORY_SCOPE_AGENT);
}

// ---------------------------------------------------------------------------
// Node pass: zero the nodal-force accumulator (region 0), copy F_ext
// (region 1), compute phys_disp (region 2). Purely bandwidth bound.
// ---------------------------------------------------------------------------
__global__ __launch_bounds__(256) void node_kernel(
    const float* __restrict__ pred,
    const float* __restrict__ fext,
    const float* __restrict__ u_scale,
    const float* __restrict__ theta_scale,
    float* __restrict__ out,
    int n3)                       // n3 = 3 * N_NODES
{
    int i = blockIdx.x * blockDim.x + threadIdx.x;
    if (i >= n3) return;
    float u  = u_scale[0];
    float th = theta_scale[0];
    out[i]          = 0.0f;
    out[n3 + i]     = fext[i];
    float p  = pred[i];
    float sc = ((i % 3) == 2) ? th : u;
    out[2 * n3 + i] = p * sc;
}

// ---------------------------------------------------------------------------
// Edge pass: double-buffered async Global->LDS staging of the streaming edge
// data (conn/E/A/I), random gathers of pred/coords (L2-resident), beam math,
// and 6 hardware FP32 atomic adds per edge into the L2-resident accumulator.
// ---------------------------------------------------------------------------
__global__ __launch_bounds__(TILE) void edge_kernel(
    const float* __restrict__ pred,     // (N,3)
    const int*   __restrict__ conn,     // (E,2) int32
    const float* __restrict__ coords,   // (N,3)
    const float* __restrict__ E_,
    const float* __restrict__ A_,
    const float* __restrict__ I_,
    float* __restrict__ forces,         // accumulator = out region 0
    int nE)
{
    __shared__ int2  s_conn[2][TILE];
    __shared__ float s_E[2][TILE];
    __shared__ float s_A[2][TILE];
    __shared__ float s_I[2][TILE];

    const int tid    = threadIdx.x;
    const int nTiles = (nE + TILE - 1) / TILE;
    const int2* conn2 = (const int2*)conn;

    int t = blockIdx.x;
    if (t >= nTiles) return;

    // Stage tile t into buffer 0. OOB lanes clamp their address so every
    // stage is exactly 4 async instructions (keeps wait counts exact).
    {
        int e = t * TILE + tid; if (e >= nE) e = nE - 1;
        async_ld_b64(&conn2[e], &s_conn[0][tid]);
        async_ld_b32(&E_[e],    &s_E[0][tid]);
        async_ld_b32(&A_[e],    &s_A[0][tid]);
        async_ld_b32(&I_[e],    &s_I[0][tid]);
    }

    int buf = 0;
    for (; t < nTiles; t += gridDim.x) {
        int tn = t + gridDim.x;
        if (tn < nTiles) {
            // Stage next tile into the other buffer, then wait until only its
            // 4 ops remain in flight => current tile's data has landed
            // (async loads complete in order).
            int e = tn * TILE + tid; if (e >= nE) e = nE - 1;
            int nb = buf ^ 1;
            async_ld_b64(&conn2[e], &s_conn[nb][tid]);
            async_ld_b32(&E_[e],    &s_E[nb][tid]);
            async_ld_b32(&A_[e],    &s_A[nb][tid]);
            async_ld_b32(&I_[e],    &s_I[nb][tid]);
            WAIT_ASYNC(4);
        } else {
            WAIT_ASYNC(0);
        }

        int e = t * TILE + tid;
        if (e < nE) {
            int2 cn = s_conn[buf][tid];          // ds_load_b64, conflict-free
            int a = cn.x;
            int b = cn.y;
            float Ee = s_E[buf][tid];
            float Ae = s_A[buf][tid];
            float Ie = s_I[buf][tid];

            const F3* predv   = (const F3*)pred;
            const F3* coordsv = (const F3*)coords;
            F3 ca = coordsv[a];
            F3 cb = coordsv[b];
            F3 dA = predv[a];
            F3 dB = predv[b];

            float dx0 = cb.x - ca.x;
            float dz0 = cb.z - ca.z;
            float l0  = sqrtf(dx0 * dx0 + dz0 * dz0 + EPSF);
            float inv = 1.0f / (l0 + EPSF);
            float c = dx0 * inv;
            float s = dz0 * inv;

            float EA = Ee * Ae;
            float EI = Ee * Ie;
            float l2 = l0 * l0;
            float l3 = l2 * l0;
            float k_ax   = EA * inv;
            float k_bend = EI * inv;
            float k_sw   = EI / (l2 + EPSF);
            float k_tr   = EI / (l3 + EPSF);

            float ua =  c * dA.x + s * dA.y;
            float wa = -s * dA.x + c * dA.y;
            float ta = -dA.z;
            float ub =  c * dB.x + s * dB.y;
            float wb = -s * dB.x + c * dB.y;
            float tb = -dB.z;

            float f0 = k_ax * (ua - ub);
            float f3 = k_ax * (ub - ua);
            float f1 = 12.0f * k_tr * (wa - wb) + 6.0f * k_sw * (ta + tb);
            float f4 = 12.0f * k_tr * (wb - wa) - 6.0f * k_sw * (ta + tb);
            float f2 = 6.0f * k_sw * (wa - wb) + k_bend * (4.0f * ta + 2.0f * tb);
            float f5 = 6.0f * k_sw * (wa - wb) + k_bend * (2.0f * ta + 4.0f * tb);

            float fAx = c * f0 - s * f1;
            float fAy = s * f0 + c * f1;
            float fBx = c * f3 - s * f4;
            float fBy = s * f3 + c * f4;

            float* oA = forces + 3 * a;
            float* oB = forces + 3 * b;
            atomic_add_f32(oA + 0, fAx);
            atomic_add_f32(oA + 1, fAy);
            atomic_add_f32(oA + 2, f2);
            atomic_add_f32(oB + 0, fBx);
            atomic_add_f32(oB + 1, fBy);
            atomic_add_f32(oB + 2, f5);
        }
        buf ^= 1;
    }
}

extern "C" void kernel_launch(void* const* d_in, const int* in_sizes, int n_in,
                              void* d_out, int out_size, void* d_ws, size_t ws_size,
                              hipStream_t stream) {
    const float* pred   = (const float*)d_in[0];
    const int*   conn   = (const int*)  d_in[1];   // harness dtype table: integer -> int32
    const float* coords = (const float*)d_in[2];
    const float* E_     = (const float*)d_in[3];
    const float* A_     = (const float*)d_in[4];
    const float* I_     = (const float*)d_in[5];
    const float* fext   = (const float*)d_in[6];
    const float* us     = (const float*)d_in[7];
    const float* ts     = (const float*)d_in[8];
    float* out = (float*)d_out;

    const int n3 = in_sizes[0];   // 3 * N_NODES
    const int nE = in_sizes[3];   // N_EDGES

    // Node pass first: zero-init the accumulator region before edge atomics.
    node_kernel<<<(n3 + 255) / 256, 256, 0, stream>>>(pred, fext, us, ts, out, n3);

    // Edge pass: 4096 blocks x 256 threads; 15625 tiles of 256 edges ->
    // ~3.8 tiles per block, so the double-buffered async pipeline has work.
    edge_kernel<<<4096, TILE, 0, stream>>>(pred, conn, coords, E_, A_, I_, out, nE);
}